// Proposal_Layer_7473243095571
// MI455X (gfx1250) — compile-verified
//
#include <hip/hip_runtime.h>
#include <math.h>

#define B_    8
#define N_    16384
#define C_    256
#define H_    128
#define NPRE  900
#define NPOST 64
#define MW    29            // ceil(900/32) suppression-mask words per row
#define NBINS 65536
#define NMS_T 0.85f
#define BN_EPS 1e-5f

typedef float v2f __attribute__((ext_vector_type(2)));
typedef float v8f __attribute__((ext_vector_type(8)));

// ---------------------------------------------------------------------------
// Kernel 0: zero the score histogram + candidate counters (ws is poisoned).
// ---------------------------------------------------------------------------
__global__ void zero_kernel(unsigned* __restrict__ hist, int* __restrict__ cnt)
{
    int g = blockIdx.x * blockDim.x + threadIdx.x;
    if (g < B_ * NBINS) hist[g] = 0u;
    if (g < B_) cnt[g] = 0;
}

// ---------------------------------------------------------------------------
// Kernel 1: fused MLP.  One block = 256 threads = 8 wave32s, one 16-point tile.
// Wave w computes output channels [16w, 16w+16) via V_WMMA_F32_16X16X4_F32
// accumulating over K=256 in steps of 4.  Then BN+ReLU -> LDS, tiny (4x128)
// layer 2, vote_xyz, prop_out, sigmoid score + histogram vote.
// ---------------------------------------------------------------------------
__global__ __launch_bounds__(256)
void mlp_wmma_kernel(const float* __restrict__ xyz,     // (B,N,3)
                     const float* __restrict__ feat,    // (B,C,N)
                     const float* __restrict__ est,     // (B,N,1)
                     const float* __restrict__ W1,      // (H,C)
                     const float* __restrict__ b1,      // (H)
                     const float* __restrict__ gamma,   // (H)
                     const float* __restrict__ beta,    // (H)
                     const float* __restrict__ rmean,   // (H)
                     const float* __restrict__ rvar,    // (H)
                     const float* __restrict__ W2,      // (4,H)
                     const float* __restrict__ b2,      // (4)
                     float* __restrict__ out_prop,      // (B,N,4)
                     float* __restrict__ ws_scores,     // (B,N)
                     unsigned* __restrict__ ws_hist)    // (B,NBINS)
{
    __shared__ float s_scale[H_];
    __shared__ float s_bias[H_];
    __shared__ float h_s[H_ * 16];   // h[ch][col]
    __shared__ float off_s[16 * 4];  // offset[col][o]

    const int tid = threadIdx.x;
    const int b   = blockIdx.y;
    const int n0  = blockIdx.x * 16;

    if (tid < H_) {
        float inv = gamma[tid] * rsqrtf(rvar[tid] + BN_EPS);
        s_scale[tid] = inv;
        s_bias[tid]  = b1[tid] * inv + beta[tid] - rmean[tid] * inv;
    }
    __syncthreads();

    const int wave = tid >> 5;       // 0..7 (wave32)
    const int lane = tid & 31;
    const int half = lane >> 4;
    const int l15  = lane & 15;

    const int arow = wave * 16 + l15;
    const int col  = n0 + l15;
    const float* featb = feat + (size_t)b * C_ * N_;

    v8f acc = {0.f, 0.f, 0.f, 0.f, 0.f, 0.f, 0.f, 0.f};

    for (int k0 = 0; k0 < C_; k0 += 4) {
        const int kb = k0 + half * 2;
        v2f a, bb;
        // A (16x4 fp32): VGPR0 = K{0|2}, VGPR1 = K{1|3}
        a.x = W1[arow * C_ + kb];
        a.y = W1[arow * C_ + kb + 1];
        // B (4x16 fp32): mirrored layout, row-striped across lanes
        bb.x = featb[(size_t)kb * N_ + col];
        bb.y = featb[(size_t)(kb + 1) * N_ + col];
        if (k0 + 4 < C_)  // locality 3 -> WGP-scope prefetch (all cache levels)
            __builtin_prefetch(&featb[(size_t)(kb + 4) * N_ + col], 0, 3);
        acc = __builtin_amdgcn_wmma_f32_16x16x4_f32(
                  false, a, false, bb, (short)0, acc, false, false);
    }

    // C/D layout: VGPR v, lanes 0-15 -> M=v ; lanes 16-31 -> M=v+8
#pragma unroll
    for (int v = 0; v < 8; ++v) {
        int ch = wave * 16 + half * 8 + v;
        float val = acc[v] * s_scale[ch] + s_bias[ch];
        h_s[ch * 16 + l15] = fmaxf(val, 0.f);
    }
    __syncthreads();

    if (tid < 64) {
        int o = tid & 3, c2 = tid >> 2;
        float a2 = b2[o];
#pragma unroll 8
        for (int ch = 0; ch < H_; ++ch)
            a2 += W2[o * H_ + ch] * h_s[ch * 16 + c2];
        off_s[c2 * 4 + o] = a2;
    }
    __syncthreads();

    if (tid < 16) {
        int n = n0 + tid;
        size_t base3 = ((size_t)b * N_ + n) * 3;
        float vx = xyz[base3 + 0] + off_s[tid * 4 + 0];
        float vy = xyz[base3 + 1] + off_s[tid * 4 + 1];
        float vz = xyz[base3 + 2] + off_s[tid * 4 + 2];
        size_t base4 = ((size_t)b * N_ + n) * 4;
        out_prop[base4 + 0] = vx;
        out_prop[base4 + 1] = vy;
        out_prop[base4 + 2] = vz;
        out_prop[base4 + 3] = off_s[tid * 4 + 3];
        float cl = est[(size_t)b * N_ + n];
        float sc = 1.f / (1.f + expf(-cl));
        ws_scores[(size_t)b * N_ + n] = sc;
        // sigmoid > 0 => IEEE bits are order-preserving as unsigned
        atomicAdd(&ws_hist[(size_t)b * NBINS + (__float_as_uint(sc) >> 16)], 1u);
    }
}

// ---------------------------------------------------------------------------
// Kernel 2: per-batch cutoff bin: highest bin B* such that
// count(bins >= B*) >= NPRE.  Elements in bins > B* surely rank < NPRE;
// elements in bins < B* surely rank >= NPRE.  Two-level top-down scan.
// ---------------------------------------------------------------------------
__global__ __launch_bounds__(256)
void cutoff_kernel(const unsigned* __restrict__ hist, int* __restrict__ cutoff)
{
    __shared__ unsigned part[256];
    const int b = blockIdx.x, tid = threadIdx.x;
    const unsigned* h = hist + (size_t)b * NBINS;
    unsigned s = 0;
#pragma unroll 8
    for (int k = 0; k < NBINS / 256; ++k) s += h[tid * (NBINS / 256) + k];
    part[tid] = s;
    __syncthreads();
    if (tid == 0) {
        unsigned cum = 0;
        int seg = 0;
        for (int sg = 255; sg >= 0; --sg) {
            if (cum + part[sg] >= NPRE) { seg = sg; break; }
            cum += part[sg];
        }
        int bin = seg * 256 + 255;
        for (; bin > seg * 256; --bin) {
            cum += h[bin];
            if (cum >= NPRE) break;
        }
        cutoff[b] = bin;
    }
}

// ---------------------------------------------------------------------------
// Kernel 3: compact candidate indices (bins >= cutoff).  Order of the
// compacted list is nondeterministic but irrelevant: output slots come from
// exact ranks.
// ---------------------------------------------------------------------------
__global__ __launch_bounds__(256)
void compact_kernel(const float* __restrict__ ws_scores,
                    const int* __restrict__ cutoff,
                    int* __restrict__ cand, int* __restrict__ cnt)
{
    const int b = blockIdx.y;
    const int i = blockIdx.x * blockDim.x + threadIdx.x;
    unsigned key = __float_as_uint(ws_scores[(size_t)b * N_ + i]);
    if ((int)(key >> 16) >= cutoff[b]) {
        int pos = atomicAdd(&cnt[b], 1);
        cand[(size_t)b * N_ + pos] = i;
    }
}

// ---------------------------------------------------------------------------
// Kernel 4: exact stable rank for candidates only.
// rank(i) = #{j: s_j > s_i} + #{j < i: s_j == s_i}  (== stable argsort(-s)).
// rank < NPRE -> scatter score + reconstructed 7-float proposal.
// ---------------------------------------------------------------------------
__global__ __launch_bounds__(256)
void rank_kernel(const float* __restrict__ ws_scores,
                 const float* __restrict__ out_prop,
                 const float* __restrict__ tbox,      // (B,1,7)
                 const int* __restrict__ cand,
                 const int* __restrict__ cnt,
                 float* __restrict__ topk_s,          // (B,NPRE)
                 float* __restrict__ topk_p)          // (B,NPRE,7)
{
    __shared__ float tile[256];
    const int b = blockIdx.y, tid = threadIdx.x;
    const int nc = cnt[b];
    if (blockIdx.x * blockDim.x >= nc) return;   // uniform early-out
    const int slot = blockIdx.x * blockDim.x + tid;
    const float* sc = ws_scores + (size_t)b * N_;

    const bool act = slot < nc;
    const int  i   = act ? cand[(size_t)b * N_ + slot] : 0;
    const float si = sc[i];

    int rank = 0;
    for (int j0 = 0; j0 < N_; j0 += 256) {
        __syncthreads();
        tile[tid] = sc[j0 + tid];
        __syncthreads();
#pragma unroll 8
        for (int t = 0; t < 256; ++t) {
            float sj = tile[t];
            int j = j0 + t;
            rank += (sj > si) || (sj == si && j < i);
        }
    }
    if (act && rank < NPRE) {
        size_t src = ((size_t)b * N_ + i) * 4;
        size_t dst = ((size_t)b * NPRE + rank) * 7;
        topk_s[(size_t)b * NPRE + rank] = si;
        topk_p[dst + 0] = out_prop[src + 0];
        topk_p[dst + 1] = out_prop[src + 1];
        topk_p[dst + 2] = out_prop[src + 2];
        topk_p[dst + 3] = tbox[b * 7 + 3];
        topk_p[dst + 4] = tbox[b * 7 + 4];
        topk_p[dst + 5] = tbox[b * 7 + 5];
        topk_p[dst + 6] = out_prop[src + 3];
    }
}

// ---------------------------------------------------------------------------
// Sutherland-Hodgman intersection area of two rotated rectangles
// (reference _pair_inter semantics, MAXV = 8).
// ---------------------------------------------------------------------------
__device__ float pair_inter(const float* __restrict__ ca,
                            const float* __restrict__ cb)
{
    float px[8], py[8];
    int cnt = 4;
#pragma unroll
    for (int k = 0; k < 4; ++k) { px[k] = ca[2 * k]; py[k] = ca[2 * k + 1]; }
#pragma unroll
    for (int k = 4; k < 8; ++k) { px[k] = 0.f; py[k] = 0.f; }

    for (int e = 0; e < 4; ++e) {
        float ax = cb[2 * e], ay = cb[2 * e + 1];
        int e2 = (e + 1) & 3;
        float nx = cb[2 * e2] - ax, ny = cb[2 * e2 + 1] - ay;
        float qx[8], qy[8];
        int ocnt = 0;
        for (int i = 0; i < 8; ++i) {
            bool valid = (i < cnt);
            float cx = px[i], cy = py[i];
            int in = (i + 1 < cnt) ? i + 1 : 0;
            float x2 = px[in], y2 = py[in];
            float dc = nx * (cy - ay) - ny * (cx - ax);
            float dn = nx * (y2 - ay) - ny * (x2 - ax);
            bool in_c = dc >= 0.f, in_n = dn >= 0.f;
            if (valid && in_c && ocnt < 8) {
                qx[ocnt] = cx; qy[ocnt] = cy; ++ocnt;
            }
            if (valid && (in_c != in_n) && ocnt < 8) {
                float denom = dc - dn;
                float t = dc / ((denom == 0.f) ? 1.f : denom);
                qx[ocnt] = cx + t * (x2 - cx);
                qy[ocnt] = cy + t * (y2 - cy);
                ++ocnt;
            }
        }
        cnt = ocnt;
        for (int i = 0; i < 8; ++i) {
            px[i] = (i < ocnt) ? qx[i] : 0.f;
            py[i] = (i < ocnt) ? qy[i] : 0.f;
        }
    }
    float s = 0.f;
    if (cnt >= 3) {
        for (int i = 0; i < cnt; ++i) {
            int j = (i + 1 < cnt) ? i + 1 : 0;
            s += px[i] * py[j] - px[j] * py[i];
        }
    }
    return fabsf(s) * 0.5f;
}

// ---------------------------------------------------------------------------
// Kernel 5: fully parallel suppression bitmasks.  Thread g = (i, word w)
// computes 32 IoUs vs boxes j in [32w, 32w+32), bit set iff j > i and
// IoU > NMS_T.  Corners built from proposals into LDS per block.
// ~816 blocks fill the chip (vs 8 sequential blocks before).
// ---------------------------------------------------------------------------
__global__ __launch_bounds__(256)
void iou_mask_kernel(const float* __restrict__ topk_p,
                     unsigned* __restrict__ masks)   // (B,NPRE,MW)
{
    __shared__ float s_cor[NPRE * 8];
    __shared__ float s_area[NPRE];
    const int b = blockIdx.y, tid = threadIdx.x, bd = blockDim.x;

    for (int i = tid; i < NPRE; i += bd) {
        const float* p = topk_p + ((size_t)b * NPRE + i) * 7;
        float cu = p[0], cv = p[2];
        float hl = p[5] * 0.5f, hw = p[4] * 0.5f;
        float c = cosf(p[6]), s = sinf(p[6]);
        const float lx[4] = {-hl, hl, hl, -hl};
        const float ly[4] = {-hw, -hw, hw, hw};
#pragma unroll
        for (int k = 0; k < 4; ++k) {
            s_cor[i * 8 + 2 * k]     = cu + c * lx[k] - s * ly[k];
            s_cor[i * 8 + 2 * k + 1] = cv + s * lx[k] + c * ly[k];
        }
        s_area[i] = (2.f * hl) * (2.f * hw);
    }
    __syncthreads();

    const int g = blockIdx.x * bd + tid;
    if (g >= NPRE * MW) return;
    const int i = g / MW;
    const int w = g % MW;

    float ci[8];
#pragma unroll
    for (int k = 0; k < 8; ++k) ci[k] = s_cor[i * 8 + k];
    const float ai = s_area[i];

    unsigned m = 0;
    for (int t = 0; t < 32; ++t) {
        int j = w * 32 + t;
        if (j >= NPRE || j <= i) continue;
        float inter = pair_inter(ci, &s_cor[j * 8]);
        float iou = inter / fmaxf(ai + s_area[j] - inter, 1e-8f);
        if (iou > NMS_T) m |= (1u << t);
    }
    masks[((size_t)b * NPRE + i) * MW + w] = m;
}

// ---------------------------------------------------------------------------
// Kernel 6: cheap sequential NMS reduce (equivalent to the reference
// fori_loop), rank kept boxes, zero-fill + scatter <=64 outputs.
// ---------------------------------------------------------------------------
__global__ __launch_bounds__(256)
void reduce_scatter_kernel(const unsigned* __restrict__ masks,
                           const float* __restrict__ topk_s,
                           const float* __restrict__ topk_p,
                           float* __restrict__ out_bbox,    // (B,64,7)
                           float* __restrict__ out_scores,  // (B,64)
                           float* __restrict__ out_center)  // (B,64,3)
{
    __shared__ unsigned rem[MW];
    __shared__ int s_rank[NPRE];
    const int b = blockIdx.x, tid = threadIdx.x, bd = blockDim.x;

    for (int k = tid; k < NPOST * 7; k += bd) out_bbox[b * NPOST * 7 + k] = 0.f;
    for (int k = tid; k < NPOST;     k += bd) out_scores[b * NPOST + k]   = 0.f;
    for (int k = tid; k < NPOST * 3; k += bd) out_center[b * NPOST * 3 + k] = 0.f;

    if (tid < MW) rem[tid] = (tid == MW - 1) ? 0xFu : 0xFFFFFFFFu;  // 900 bits
    __syncthreads();

    for (int i = 0; i < NPRE; ++i) {
        bool kept = (rem[i >> 5] >> (i & 31)) & 1u;
        __syncthreads();              // all read before any write
        if (kept && tid < MW)
            rem[tid] &= ~masks[((size_t)b * NPRE + i) * MW + tid];
        __syncthreads();              // writes visible before next read
    }

    if (tid == 0) {
        int c = 0;
        for (int i = 0; i < NPRE; ++i) {
            bool kept = (rem[i >> 5] >> (i & 31)) & 1u;
            s_rank[i] = kept ? c : NPOST;
            c += kept;
        }
    }
    __syncthreads();

    for (int i = tid; i < NPRE; i += bd) {
        int r = s_rank[i];
        if (r < NPOST) {
            const float* p = topk_p + ((size_t)b * NPRE + i) * 7;
#pragma unroll
            for (int k = 0; k < 7; ++k)
                out_bbox[((size_t)b * NPOST + r) * 7 + k] = p[k];
            out_scores[b * NPOST + r] = topk_s[(size_t)b * NPRE + i];
#pragma unroll
            for (int k = 0; k < 3; ++k)
                out_center[((size_t)b * NPOST + r) * 3 + k] = p[k];
        }
    }
}

// ---------------------------------------------------------------------------
extern "C" void kernel_launch(void* const* d_in, const int* in_sizes, int n_in,
                              void* d_out, int out_size, void* d_ws, size_t ws_size,
                              hipStream_t stream)
{
    const float* xyz   = (const float*)d_in[0];
    const float* feat  = (const float*)d_in[1];
    const float* est   = (const float*)d_in[2];
    const float* tbox  = (const float*)d_in[3];
    const float* W1    = (const float*)d_in[4];
    const float* b1    = (const float*)d_in[5];
    const float* gamma = (const float*)d_in[6];
    const float* beta  = (const float*)d_in[7];
    const float* rmean = (const float*)d_in[8];
    const float* rvar  = (const float*)d_in[9];
    const float* W2    = (const float*)d_in[10];
    const float* b2    = (const float*)d_in[11];

    float* out        = (float*)d_out;
    float* out_bbox   = out;                                   // 8*64*7
    float* out_scores = out + (size_t)B_ * NPOST * 7;          // 8*64
    float* out_prop   = out_scores + (size_t)B_ * NPOST;       // 8*16384*4
    float* out_center = out_prop + (size_t)B_ * N_ * 4;        // 8*64*3

    // workspace layout (all 4-byte elements)
    float*    ws        = (float*)d_ws;
    float*    ws_scores = ws;                                  // B*N
    float*    ws_topk_s = ws_scores + (size_t)B_ * N_;         // B*NPRE
    float*    ws_topk_p = ws_topk_s + (size_t)B_ * NPRE;       // B*NPRE*7
    unsigned* ws_hist   = (unsigned*)(ws_topk_p + (size_t)B_ * NPRE * 7); // B*NBINS
    unsigned* ws_masks  = ws_hist + (size_t)B_ * NBINS;        // B*NPRE*MW
    int*      ws_cand   = (int*)(ws_masks + (size_t)B_ * NPRE * MW); // B*N
    int*      ws_cnt    = ws_cand + (size_t)B_ * N_;           // B
    int*      ws_cutoff = ws_cnt + B_;                         // B

    zero_kernel<<<(B_ * NBINS + 255) / 256, 256, 0, stream>>>(ws_hist, ws_cnt);

    mlp_wmma_kernel<<<dim3(N_ / 16, B_), 256, 0, stream>>>(
        xyz, feat, est, W1, b1, gamma, beta, rmean, rvar, W2, b2,
        out_prop, ws_scores, ws_hist);

    cutoff_kernel<<<B_, 256, 0, stream>>>(ws_hist, ws_cutoff);

    compact_kernel<<<dim3(N_ / 256, B_), 256, 0, stream>>>(
        ws_scores, ws_cutoff, ws_cand, ws_cnt);

    rank_kernel<<<dim3(N_ / 256, B_), 256, 0, stream>>>(
        ws_scores, out_prop, tbox, ws_cand, ws_cnt, ws_topk_s, ws_topk_p);

    iou_mask_kernel<<<dim3((NPRE * MW + 255) / 256, B_), 256, 0, stream>>>(
        ws_topk_p, ws_masks);

    reduce_scatter_kernel<<<B_, 256, 0, stream>>>(
        ws_masks, ws_topk_s, ws_topk_p, out_bbox, out_scores, out_center);
}